// pi_GNN_78331613545121
// MI455X (gfx1250) — compile-verified
//
#include <hip/hip_runtime.h>
#include <hip/hip_bf16.h>
#include <math.h>

// ---------------- problem constants ----------------
#define BATCH 2048
#define NM    128      // N_MAX
#define IDIM  32       // INPUT_DIM
#define HDIM  64       // HIDDEN_DIM
#define HN    32
#define MZ    33       // HN+1
#define NZ    129      // NM+1
#define ZS    132      // Z row stride (floats)
#define PS    132      // Pc / A1 row stride (banks: 4l pattern, conflict-free)
#define FS    36       // feats row stride  (banks: 36l mod 64, all distinct)
#define HS    68       // h-scratch row stride (banks: 4l pattern)
#define ITERS 100
#define LNEPS 1e-5f

typedef __attribute__((ext_vector_type(2))) float v2f;
typedef __attribute__((ext_vector_type(8))) float v8f;

#define WMMA4(a, b, c) __builtin_amdgcn_wmma_f32_16x16x4_f32(false, (a), false, (b), (short)0, (c), false, false)

// =====================================================================
// Kernel 1: per-graph fused pipeline, one 256-thread block (8 wave32) per
// graph. All matmuls on V_WMMA_F32_16X16X4_F32; Sinkhorn on VALU/LDS.
// =====================================================================
__global__ __launch_bounds__(256)
void k_graph(const float* __restrict__ x, const float* __restrict__ adj,
             const float* __restrict__ weight,
             const float* __restrict__ w1, const float* __restrict__ b1,
             const float* __restrict__ ln1g, const float* __restrict__ ln1b,
             const float* __restrict__ ln2g, const float* __restrict__ ln2b,
             const float* __restrict__ bin,
             float* __restrict__ A_n, float* __restrict__ F_n)
{
    __shared__ float s_feats[NM * FS];     // 18.4 KB  feats, padded stride
    __shared__ float s_hp[64 * HS];        // 17.4 KB  h-half early; Pc later
    __shared__ float s_Z[MZ * ZS];         // 17.4 KB  Z; A1 later
    __shared__ float s_u[MZ];
    __shared__ float s_v[NZ];
    __shared__ float s_pm[256];            // v-pass partial max
    __shared__ float s_ps[256];            // v-pass partial sum
    __shared__ float s_red[8];
    __shared__ float s_red2[8];

    const int g    = blockIdx.x;
    const int tid  = threadIdx.x;
    const int lane = tid & 31;
    const int wave = tid >> 5;
    const int hf   = lane >> 4;            // 0: K=0,1   1: K=2,3 (frag half)
    const int l    = lane & 15;

    const float norm   = -logf(160.0f);    // -log(m+n), m=32 n=128
    const float LOGN   = logf(128.0f);
    const float LOGM   = logf(32.0f);
    const float NEGINF = -1.0e30f;

    // ---- prefetch this graph's adj into L2/WGP$ (consumed much later by
    //      the Pc@adj WMMA loop; Sinkhorn gives it plenty of time to land)
    {
        const char* ap = (const char*)(adj + (size_t)g * (NM * NM));
        for (int off = tid * 128; off < NM * NM * 4; off += 256 * 128)
            __builtin_prefetch(ap + off, 0, 2);
    }

    // ---- load feats tile (padded stride FS) ----
    const float* xg = x + (size_t)g * (NM * IDIM);
    for (int i = tid; i < NM * IDIM; i += 256) {
        int n = i >> 5, d = i & 31;
        s_feats[n * FS + d] = xg[i];
    }
    __syncthreads();

    // =================================================================
    // h = relu(feats@w1+b1), scores = relu(weight@h^T) -> Z interior.
    // Processed in two n-halves of 64 so h fits in s_hp. All WMMA.
    // =================================================================
    for (int hb = 0; hb < 2; ++hb) {
        const int n0 = hb * 64;
        // ---- h tiles: (64 x 64), K=32: 4x4 = 16 tiles, 2 per wave ----
        for (int t = wave; t < 16; t += 8) {
            const int mt = t >> 2, nt = t & 3;
            v8f c = {};
            for (int k = 0; k < IDIM; k += 4) {
                v2f a, b;
                a.x = s_feats[(n0 + mt * 16 + l) * FS + k + 2 * hf];
                a.y = s_feats[(n0 + mt * 16 + l) * FS + k + 2 * hf + 1];
                b.x = w1[(k + 2 * hf) * HDIM + nt * 16 + l];
                b.y = w1[(k + 2 * hf + 1) * HDIM + nt * 16 + l];
                c = WMMA4(a, b, c);
            }
            const int rl  = mt * 16 + 8 * hf;
            const int col = nt * 16 + l;
            const float bb = b1[col];
#pragma unroll
            for (int r = 0; r < 8; ++r)
                s_hp[(rl + r) * HS + col] = fmaxf(c[r] + bb, 0.0f);
        }
        __syncthreads();
        // ---- scores tiles: (32 x 64), K=64: 2x4 = 8 tiles, 1 per wave ----
        {
            const int mt = wave & 1, nt = wave >> 1;
            v8f c = {};
            for (int k = 0; k < HDIM; k += 4) {
                v2f a, b;
                a.x = weight[(mt * 16 + l) * HDIM + k + 2 * hf];
                a.y = weight[(mt * 16 + l) * HDIM + k + 2 * hf + 1];
                b.x = s_hp[(nt * 16 + l) * HS + k + 2 * hf];     // h^T frag
                b.y = s_hp[(nt * 16 + l) * HS + k + 2 * hf + 1];
                c = WMMA4(a, b, c);
            }
            const int d0 = mt * 16 + 8 * hf;
#pragma unroll
            for (int r = 0; r < 8; ++r)
                s_Z[(d0 + r) * ZS + n0 + nt * 16 + l] = fmaxf(c[r], 0.0f);
        }
        __syncthreads();
    }

    // ---- Z borders, u/v init ----
    const float alpha = bin[0];
    for (int i = tid; i < NZ; i += 256) s_Z[32 * ZS + i] = alpha;   // bins1 row + corner
    if (tid < MZ)  s_Z[tid * ZS + 128] = alpha;                      // bins0 column
    if (tid < MZ)  s_u[tid] = 0.0f;
    if (tid < NZ)  s_v[tid] = 0.0f;
    __syncthreads();

    // ---- Sinkhorn, 100 iterations (log-domain) ----
    for (int it = 0; it < ITERS; ++it) {
        // u[m] = log_mu[m] - logsumexp_n(Z[m][n]+v[n]); rows over waves.
        // Z+v cached in registers between the max pass and the sum pass.
        for (int m = wave; m < MZ; m += 8) {
            float zv[5];
            float mx = NEGINF;
#pragma unroll
            for (int i = 0; i < 5; ++i) {
                int n = lane + 32 * i;
                if (n < NZ) {
                    zv[i] = s_Z[m * ZS + n] + s_v[n];
                    mx = fmaxf(mx, zv[i]);
                }
            }
#pragma unroll
            for (int o = 16; o > 0; o >>= 1) mx = fmaxf(mx, __shfl_xor(mx, o, 32));
            float sm = 0.0f;
#pragma unroll
            for (int i = 0; i < 5; ++i) {
                int n = lane + 32 * i;
                if (n < NZ) sm += __expf(zv[i] - mx);
            }
#pragma unroll
            for (int o = 16; o > 0; o >>= 1) sm += __shfl_xor(sm, o, 32);
            if (lane == 0) {
                float lmu = (m < HN) ? norm : (norm + LOGN);
                s_u[m] = lmu - (mx + __logf(sm));
            }
        }
        __syncthreads();

        // v[n] = log_nu[n] - logsumexp_m(Z[m][n]+u[m]).
        // Phase 1: 2-way row split over all 256 threads for cols 0..127.
        {
            const int col  = tid & 127;
            const int part = tid >> 7;           // uniform per wave
            const int m0p  = part ? 17 : 0;
            const int nr   = part ? 16 : 17;
            float zv[17];
            float mx = NEGINF;
#pragma unroll
            for (int i = 0; i < 17; ++i) {
                if (i < nr) {
                    zv[i] = s_Z[(m0p + i) * ZS + col] + s_u[m0p + i];
                    mx = fmaxf(mx, zv[i]);
                }
            }
            float sm = 0.0f;
#pragma unroll
            for (int i = 0; i < 17; ++i)
                if (i < nr) sm += __expf(zv[i] - mx);
            s_pm[tid] = mx;
            s_ps[tid] = sm;
        }
        __syncthreads();
        // Phase 2: logsumexp-merge for cols 0..127; col 128 (constant-alpha
        // column of Z) collapses to alpha + logsumexp(u), done by wave 4.
        if (tid < 128) {
            const float m1v = s_pm[tid],       s1v = s_ps[tid];
            const float m2v = s_pm[128 + tid], s2v = s_ps[128 + tid];
            const float M = fmaxf(m1v, m2v);
            const float S = s1v * __expf(m1v - M) + s2v * __expf(m2v - M);
            s_v[tid] = norm - (M + __logf(S));
        } else if (tid < 160) {                  // wave 4, lane-parallel over u
            const int ln = tid - 128;
            float a1 = s_u[ln];
            float a2 = (ln == 0) ? s_u[32] : NEGINF;
            float mx = fmaxf(a1, a2);
#pragma unroll
            for (int o = 16; o > 0; o >>= 1) mx = fmaxf(mx, __shfl_xor(mx, o, 32));
            float sm = __expf(a1 - mx) + ((ln == 0) ? __expf(a2 - mx) : 0.0f);
#pragma unroll
            for (int o = 16; o > 0; o >>= 1) sm += __shfl_xor(sm, o, 32);
            if (ln == 0) s_v[128] = (norm + LOGM) - (alpha + mx + __logf(sm));
        }
        __syncthreads();
    }

    // ---- Pc = exp(Z + u + v - norm), 32x128, into s_hp region ----
    float* s_Pc = s_hp;                    // h-scratch dead; reuse (stride PS)
    for (int i = tid; i < HN * NM; i += 256) {
        int m = i >> 7, n = i & 127;
        s_Pc[m * PS + n] = __expf(s_Z[m * ZS + n] + s_u[m] + s_v[n] - norm);
    }
    __syncthreads();

    // ---- A1 = Pc @ adj (32x128, K=128), WMMA; Z region reused as A1 ----
    float* s_A1 = s_Z;
    {
        const int mt  = wave & 1;
        const int ntb = wave >> 1;                    // 0..3
        const float* adjg = adj + (size_t)g * (NM * NM);
#pragma unroll
        for (int rep = 0; rep < 2; ++rep) {
            const int nt = ntb + 4 * rep;
            v8f c = {};
            for (int k = 0; k < NM; k += 4) {
                v2f a, b;
                a.x = s_Pc[(mt * 16 + l) * PS + k + 2 * hf];
                a.y = s_Pc[(mt * 16 + l) * PS + k + 2 * hf + 1];
                b.x = adjg[(k + 2 * hf) * NM + nt * 16 + l];
                b.y = adjg[(k + 2 * hf + 1) * NM + nt * 16 + l];
                c = WMMA4(a, b, c);
            }
            const int row = mt * 16 + 8 * hf;
#pragma unroll
            for (int r = 0; r < 8; ++r)
                s_A1[(row + r) * PS + nt * 16 + l] = c[r];
        }
    }
    __syncthreads();

    // =================================================================
    // A2 = A1 @ Pc^T (32x32) on waves 0-3; Fa = Pc @ feats (32x32) on
    // waves 4-7 (wave-uniform branch, EXEC all-ones). Then whole-matrix
    // LayerNorm from the C fragments and store to global.
    // =================================================================
    {
        const int wt = wave & 3;
        const int mt = wt >> 1, nt = wt & 1;
        v8f c = {};
        if (wave < 4) {                     // A2[m][j] = sum_k A1[m][k]*Pc[j][k]
            for (int k = 0; k < NM; k += 4) {
                v2f a, b;
                a.x = s_A1[(mt * 16 + l) * PS + k + 2 * hf];
                a.y = s_A1[(mt * 16 + l) * PS + k + 2 * hf + 1];
                b.x = s_Pc[(nt * 16 + l) * PS + k + 2 * hf];     // Pc^T frag
                b.y = s_Pc[(nt * 16 + l) * PS + k + 2 * hf + 1];
                c = WMMA4(a, b, c);
            }
        } else {                            // Fa[m][d] = sum_n Pc[m][n]*feats[n][d]
            for (int k = 0; k < NM; k += 4) {
                v2f a, b;
                a.x = s_Pc[(mt * 16 + l) * PS + k + 2 * hf];
                a.y = s_Pc[(mt * 16 + l) * PS + k + 2 * hf + 1];
                b.x = s_feats[(k + 2 * hf) * FS + nt * 16 + l];
                b.y = s_feats[(k + 2 * hf + 1) * FS + nt * 16 + l];
                c = WMMA4(a, b, c);
            }
        }
        // per-wave partial sums for LayerNorm statistics
        float s1 = 0.0f, s2 = 0.0f;
#pragma unroll
        for (int r = 0; r < 8; ++r) { s1 += c[r]; s2 += c[r] * c[r]; }
#pragma unroll
        for (int o = 16; o > 0; o >>= 1) {
            s1 += __shfl_xor(s1, o, 32);
            s2 += __shfl_xor(s2, o, 32);
        }
        if (lane == 0) { s_red[wave] = s1; s_red2[wave] = s2; }
        __syncthreads();
        const int qb = (wave < 4) ? 0 : 4;
        const float S1 = s_red[qb] + s_red[qb + 1] + s_red[qb + 2] + s_red[qb + 3];
        const float S2 = s_red2[qb] + s_red2[qb + 1] + s_red2[qb + 2] + s_red2[qb + 3];
        const float mu  = S1 * (1.0f / 1024.0f);
        const float var = S2 * (1.0f / 1024.0f) - mu * mu;
        const float inv = rsqrtf(var + LNEPS);
        const float* lg = (wave < 4) ? ln1g : ln2g;
        const float* lb = (wave < 4) ? ln1b : ln2b;
        float* out = ((wave < 4) ? A_n : F_n) + (size_t)g * 1024;
        const int m0 = mt * 16 + 8 * hf;
        const int j  = nt * 16 + l;
#pragma unroll
        for (int r = 0; r < 8; ++r) {
            const int o = (m0 + r) * 32 + j;
            out[o] = (c[r] - mu) * inv * lg[o] + lb[o];
        }
    }
}

// =====================================================================
// Kernel 2: fp32 WMMA GEMM  C[M,N] = act(A[M,K] @ W[K,N] + bias)
// wave-per-16x16 tile, V_WMMA_F32_16X16X4_F32, K in steps of 4.
// =====================================================================
__global__ __launch_bounds__(256)
void k_gemm_wmma(const float* __restrict__ A, const float* __restrict__ W,
                 const float* __restrict__ bias, float* __restrict__ C,
                 int M, int K, int N, int ldc, int relu)
{
    const int lane = threadIdx.x & 31;
    const int wave = threadIdx.x >> 5;
    const int tile = blockIdx.x * 8 + wave;
    const int ntn  = N >> 4;
    const int mt   = tile / ntn;
    const int nt   = tile - mt * ntn;
    if (mt * 16 >= M) return;                    // wave-uniform: EXEC all-ones inside

    const int hf = lane >> 4;
    const int l  = lane & 15;

    const float* Arow = A + (size_t)(mt * 16 + l) * K + 2 * hf;
    const float* Wcol = W + (size_t)(2 * hf) * N + nt * 16 + l;

    v8f c = {};
    for (int k = 0; k < K; k += 4) {
        v2f a = *(const v2f*)(Arow + k);          // A[m][k+2h], A[m][k+2h+1]
        v2f b;
        b.x = Wcol[(size_t)k * N];                // W[k+2h  ][n]
        b.y = Wcol[(size_t)(k + 1) * N];          // W[k+2h+1][n]
        c = WMMA4(a, b, c);
    }

    const int col  = nt * 16 + l;
    const float bv = bias[col];
    const int row0 = mt * 16 + 8 * hf;
#pragma unroll
    for (int r = 0; r < 8; ++r) {
        float v = c[r] + bv;
        if (relu) v = fmaxf(v, 0.0f);
        C[(size_t)(row0 + r) * ldc + col] = v;
    }
}

// =====================================================================
// Kernel 3: head: logits = Y6[g,:64] @ w7(64x10) + b7 ; log_softmax
// one wave32 per row
// =====================================================================
__global__ __launch_bounds__(32)
void k_head(const float* __restrict__ Y6, const float* __restrict__ w7,
            const float* __restrict__ b7, float* __restrict__ out)
{
    const int g    = blockIdx.x;
    const int lane = threadIdx.x;
    const float x0 = Y6[(size_t)g * 64 + lane];
    const float x1 = Y6[(size_t)g * 64 + 32 + lane];

    float logits[10];
#pragma unroll
    for (int cc = 0; cc < 10; ++cc) {
        float p = x0 * w7[lane * 10 + cc] + x1 * w7[(lane + 32) * 10 + cc];
#pragma unroll
        for (int o = 16; o > 0; o >>= 1) p += __shfl_xor(p, o, 32);
        logits[cc] = p + b7[cc];                 // identical across lanes
    }
    float mx = logits[0];
#pragma unroll
    for (int cc = 1; cc < 10; ++cc) mx = fmaxf(mx, logits[cc]);
    float s = 0.0f;
#pragma unroll
    for (int cc = 0; cc < 10; ++cc) s += __expf(logits[cc] - mx);
    float lse = mx + __logf(s);

    float mine = 0.0f;
#pragma unroll
    for (int cc = 0; cc < 10; ++cc) if (lane == cc) mine = logits[cc];
    if (lane < 10) out[(size_t)g * 10 + lane] = mine - lse;
}

// =====================================================================
extern "C" void kernel_launch(void* const* d_in, const int* in_sizes, int n_in,
                              void* d_out, int out_size, void* d_ws, size_t ws_size,
                              hipStream_t stream)
{
    const float* x      = (const float*)d_in[0];
    const float* adj    = (const float*)d_in[1];
    const float* weight = (const float*)d_in[2];
    const float* w1     = (const float*)d_in[3];
    const float* b1     = (const float*)d_in[4];
    const float* ln1g   = (const float*)d_in[5];
    const float* ln1b   = (const float*)d_in[6];
    const float* w2     = (const float*)d_in[7];
    const float* b2     = (const float*)d_in[8];
    const float* w3     = (const float*)d_in[9];
    const float* b3     = (const float*)d_in[10];
    const float* ln2g   = (const float*)d_in[11];
    const float* ln2b   = (const float*)d_in[12];
    const float* w4     = (const float*)d_in[13];
    const float* b4     = (const float*)d_in[14];
    const float* w5     = (const float*)d_in[15];
    const float* b5     = (const float*)d_in[16];
    const float* w6     = (const float*)d_in[17];
    const float* b6     = (const float*)d_in[18];
    const float* w7     = (const float*)d_in[19];
    const float* b7     = (const float*)d_in[20];
    const float* bin    = (const float*)d_in[21];

    float* ws   = (float*)d_ws;
    float* A_n  = ws;                                  // 2048*1024
    float* F_n  = A_n  + (size_t)BATCH * 1024;         // 2048*1024
    float* Y2a  = F_n  + (size_t)BATCH * 1024;         // 2048*256
    float* Y2f  = Y2a  + (size_t)BATCH * 256;          // 2048*256
    float* Ycat = Y2f  + (size_t)BATCH * 256;          // 2048*256 (adj|feat halves)
    float* Y6   = Ycat + (size_t)BATCH * 256;          // 2048*64

    // per-graph Sinkhorn + graph matmuls (all WMMA) + layernorm
    k_graph<<<BATCH, 256, 0, stream>>>(x, adj, weight, w1, b1,
                                       ln1g, ln1b, ln2g, ln2b, bin, A_n, F_n);

    // adj path: 2048x1024 @ 1024x256 -> relu   (2048 tiles -> 256 blocks)
    k_gemm_wmma<<<256, 256, 0, stream>>>(A_n, w2, b2, Y2a, BATCH, 1024, 256, 256, 1);
    // feat path: same shape
    k_gemm_wmma<<<256, 256, 0, stream>>>(F_n, w4, b4, Y2f, BATCH, 1024, 256, 256, 1);
    // second layers: 2048x256 @ 256x128 -> relu, written into concat halves
    k_gemm_wmma<<<128, 256, 0, stream>>>(Y2a, w3, b3, Ycat,       BATCH, 256, 128, 256, 1);
    k_gemm_wmma<<<128, 256, 0, stream>>>(Y2f, w5, b5, Ycat + 128, BATCH, 256, 128, 256, 1);
    // fuse: 2048x256 @ 256x64 -> relu
    k_gemm_wmma<<<64, 256, 0, stream>>>(Ycat, w6, b6, Y6, BATCH, 256, 64, 64, 1);
    // head + log_softmax
    k_head<<<BATCH, 32, 0, stream>>>(Y6, w7, b7, (float*)d_out);
}